// IDNN_15178414424664
// MI455X (gfx1250) — compile-verified
//
#include <hip/hip_runtime.h>

// IDNN / PINN-style fused kernel for gfx1250 (MI455X).
// y = MLP([x1,x4]); dy = dMLP([x2,x4])/dx2 (backprop through ReLU masks);
// ddy == 0 exactly (ReLU net is piecewise linear; JAX hessian of relu chain is 0).
// All matmuls use V_WMMA_F32_16X16X4_F32 (fp32 in/out, matches reference dtype).
//
// v2 changes vs v1:
//  - Grad-path activations for layers 1/2 persist in dedicated LDS buffers
//    (SG1/SG2) and masks are re-read from LDS in the backward pass; only the
//    layer-3 mask stays in VGPRs. Cuts ~64 live VGPRs -> no s_set_vgpr_msb,
//    fewer v_dual_mov shuffles, better occupancy.
//  - ReLU via v_med3_f32(x, 0, +inf): 1 VALU op/element instead of 2.
//  - y-path staging collapsed to a single buffer SA (per-wave LDS is in-order;
//    each layer's reads precede its writes in program order).

typedef float v2f __attribute__((ext_vector_type(2)));
typedef float v4f __attribute__((ext_vector_type(4)));
typedef float v8f __attribute__((ext_vector_type(8)));

#define WAVES_PER_BLOCK 2
#define WGT_FLOATS 8976                 // weights + biases region (padded)
#define WAVE_FLOATS 3072                // SA + SG1 + SG2 (3 x 1024)

static __device__ __forceinline__ v8f wmma4(v2f a, v2f b, v8f c) {
  // D = A(16x4,f32) * B(4x16,f32) + C(16x16,f32)
  return __builtin_amdgcn_wmma_f32_16x16x4_f32(
      /*neg_a=*/false, a, /*neg_b=*/false, b,
      /*c_mod=*/(short)0, c, /*reuse_a=*/false, /*reuse_b=*/false);
}

// A chunk (16x4) from column-major staging S[64][16]:
// lane (col,half): m=col, k = 4*kc + 2*half -> (a[m][k], a[m][k+1])
static __device__ __forceinline__ v2f ldA(const float* S, int col, int half, int kc) {
  const int k = kc * 4 + half * 2;
  v2f a;
  a[0] = S[k * 16 + col];
  a[1] = S[(k + 1) * 16 + col];
  return a;
}

// Forward B tile (4x16) from row-major weights W[K][64]: B[k][n] = W[k][n]
static __device__ __forceinline__ v2f ldBfwd(const float* W, int col, int half, int nt, int kc) {
  const int n = nt * 16 + col;
  const int k = kc * 4 + half * 2;
  v2f b;
  b[0] = W[k * 64 + n];
  b[1] = W[(k + 1) * 64 + n];
  return b;
}

// Backward B tile = W^T: B[k][n] = W[n][k] -> contiguous ds_load_b64
static __device__ __forceinline__ v2f ldBtr(const float* W, int col, int half, int nt, int kc) {
  const int n = nt * 16 + col;
  const int k = kc * 4 + half * 2;
  return *(const v2f*)(W + n * 64 + k);
}

// Store a 16x16 C tile into column-major staging S[64][16] (two b128 stores/lane)
static __device__ __forceinline__ void stC(float* S, int col, int half, int nt, v8f c) {
  float* p = S + (nt * 16 + col) * 16 + half * 8;
  v4f lo = {c[0], c[1], c[2], c[3]};
  v4f hi = {c[4], c[5], c[6], c[7]};
  *(v4f*)(p) = lo;
  *(v4f*)(p + 4) = hi;
}

// Re-load a C tile from column-major staging (two b128 loads/lane)
static __device__ __forceinline__ v8f ldC(const float* S, int col, int half, int nt) {
  const float* p = S + (nt * 16 + col) * 16 + half * 8;
  v4f lo = *(const v4f*)(p);
  v4f hi = *(const v4f*)(p + 4);
  v8f r = {lo[0], lo[1], lo[2], lo[3], hi[0], hi[1], hi[2], hi[3]};
  return r;
}

// ReLU as a single v_med3_f32 per element: median(x, 0, +inf) == max(x, 0)
static __device__ __forceinline__ v8f relu8(v8f c) {
#pragma unroll
  for (int i = 0; i < 8; ++i)
    c[i] = __builtin_amdgcn_fmed3f(c[i], 0.0f, __builtin_inff());
  return c;
}

// Dual-path dense layer (K=64): shares every weight B-tile load between the
// y-path (cy) and grad-path (cg) WMMAs.
static __device__ __forceinline__ void layer64_dual(
    const float* SAin, const float* SGin, const float* WS, const float* bS,
    int col, int half, v8f cy[4], v8f cg[4]) {
#pragma unroll
  for (int nt = 0; nt < 4; ++nt) {
    float bb = bS[nt * 16 + col];
    v8f c = {bb, bb, bb, bb, bb, bb, bb, bb};
    cy[nt] = c;
    cg[nt] = c;
  }
#pragma unroll
  for (int kc = 0; kc < 16; ++kc) {
    v2f ay = ldA(SAin, col, half, kc);
    v2f ag = ldA(SGin, col, half, kc);
#pragma unroll
    for (int nt = 0; nt < 4; ++nt) {
      v2f bt = ldBfwd(WS, col, half, nt, kc);
      cy[nt] = wmma4(ay, bt, cy[nt]);
      cg[nt] = wmma4(ag, bt, cg[nt]);
    }
  }
}

// Backward layer: g_out = g_in @ W^T (mask applied by caller)
static __device__ __forceinline__ void layerT(
    const float* Sin, const float* WS, int col, int half, v8f cg[4]) {
  v8f z = {0, 0, 0, 0, 0, 0, 0, 0};
#pragma unroll
  for (int nt = 0; nt < 4; ++nt) cg[nt] = z;
#pragma unroll
  for (int kc = 0; kc < 16; ++kc) {
    v2f a = ldA(Sin, col, half, kc);
#pragma unroll
    for (int nt = 0; nt < 4; ++nt) {
      v2f bt = ldBtr(WS, col, half, nt, kc);
      cg[nt] = wmma4(a, bt, cg[nt]);
    }
  }
}

__global__ __launch_bounds__(WAVES_PER_BLOCK * 32) void idnn_pinn_kernel(
    const float* __restrict__ x1, const float* __restrict__ x2,
    const float* __restrict__ x4,
    const float* __restrict__ W1g, const float* __restrict__ b1g,
    const float* __restrict__ W2g, const float* __restrict__ b2g,
    const float* __restrict__ W3g, const float* __restrict__ b3g,
    const float* __restrict__ Wfg, const float* __restrict__ bfg,
    float* __restrict__ outY, float* __restrict__ outDY,
    float* __restrict__ outDDY) {
  // LDS: weights (shared per WG) + three 4KB buffers per wave (SA, SG1, SG2)
  __shared__ __align__(16) float lds[WGT_FLOATS + WAVES_PER_BLOCK * WAVE_FLOATS];
  float* W1S = lds;          // [8][64]
  float* W2S = lds + 512;    // [64][64]
  float* W3S = lds + 4608;   // [64][64]
  float* WfS = lds + 8704;   // [64]
  float* b1S = lds + 8768;   // [64]
  float* b2S = lds + 8832;   // [64]
  float* b3S = lds + 8896;   // [64]
  float* bfS = lds + 8960;   // [1]

  const int tid = threadIdx.x;
  const int nthr = WAVES_PER_BLOCK * 32;
  for (int i = tid; i < 512; i += nthr) W1S[i] = W1g[i];
  for (int i = tid; i < 4096; i += nthr) W2S[i] = W2g[i];
  for (int i = tid; i < 4096; i += nthr) W3S[i] = W3g[i];
  for (int i = tid; i < 64; i += nthr) {
    WfS[i] = Wfg[i];
    b1S[i] = b1g[i];
    b2S[i] = b2g[i];
    b3S[i] = b3g[i];
  }
  if (tid == 0) bfS[0] = bfg[0];
  __syncthreads();

  const int wave = tid >> 5;
  const int lane = tid & 31;
  const int col = lane & 15;   // N (or M for A-loads)
  const int half = lane >> 4;  // lane group
  float* SA = lds + WGT_FLOATS + wave * WAVE_FLOATS;  // y-path / scratch staging
  float* SG1 = SA + 1024;  // grad act layer 1 (persists to g1 mask)
  float* SG2 = SA + 2048;  // grad act layer 2 (persists to g2 mask)
  const long sbase = (long)(blockIdx.x * WAVES_PER_BLOCK + wave) * 16;

  v8f cy[4], cg[4], a3m[4];

  // ---------- Layer 1 (K=8: chunk0 = x, chunk1 = x4) ----------
  v2f aY = *(const v2f*)(x1 + (sbase + col) * 4 + half * 2);
  v2f aG = *(const v2f*)(x2 + (sbase + col) * 4 + half * 2);
  v2f aT = *(const v2f*)(x4 + (sbase + col) * 4 + half * 2);
#pragma unroll
  for (int nt = 0; nt < 4; ++nt) {
    float bb = b1S[nt * 16 + col];
    v8f c = {bb, bb, bb, bb, bb, bb, bb, bb};
    cy[nt] = c;
    cg[nt] = c;
  }
#pragma unroll
  for (int kc = 0; kc < 2; ++kc) {
    v2f ay = kc ? aT : aY;
    v2f ag = kc ? aT : aG;
#pragma unroll
    for (int nt = 0; nt < 4; ++nt) {
      v2f bt = ldBfwd(W1S, col, half, nt, kc);
      cy[nt] = wmma4(ay, bt, cy[nt]);
      cg[nt] = wmma4(ag, bt, cg[nt]);
    }
  }
#pragma unroll
  for (int nt = 0; nt < 4; ++nt) {
    stC(SA, col, half, nt, relu8(cy[nt]));
    stC(SG1, col, half, nt, relu8(cg[nt]));
  }

  // ---------- Layer 2 (reads SA/SG1, writes SA/SG2) ----------
  layer64_dual(SA, SG1, W2S, b2S, col, half, cy, cg);
#pragma unroll
  for (int nt = 0; nt < 4; ++nt) {
    stC(SA, col, half, nt, relu8(cy[nt]));
    stC(SG2, col, half, nt, relu8(cg[nt]));
  }

  // ---------- Layer 3 (reads SA/SG2, writes SA; grad act kept in regs) ------
  layer64_dual(SA, SG2, W3S, b3S, col, half, cy, cg);
#pragma unroll
  for (int nt = 0; nt < 4; ++nt) {
    a3m[nt] = relu8(cg[nt]);
    stC(SA, col, half, nt, relu8(cy[nt]));
  }

  // ---------- y = a3_y @ Wf + bf  (WMMA with B nonzero only in col 0) ----------
  v8f yc = {0, 0, 0, 0, 0, 0, 0, 0};
#pragma unroll
  for (int kc = 0; kc < 16; ++kc) {
    v2f a = ldA(SA, col, half, kc);
    const int k = kc * 4 + half * 2;
    float w0 = WfS[k], w1 = WfS[k + 1];
    v2f bt;
    bt[0] = (col == 0) ? w0 : 0.0f;
    bt[1] = (col == 0) ? w1 : 0.0f;
    yc = wmma4(a, bt, yc);
  }
  if (col == 0) {
#pragma unroll
    for (int r = 0; r < 8; ++r) SA[half * 8 + r] = yc[r];
  }
  if (lane < 16) outY[sbase + lane] = SA[lane] + bfS[0];

  // ---------- g3 = Wf * (a3_g > 0)  (from registers) ----------
#pragma unroll
  for (int nt = 0; nt < 4; ++nt) {
    float wf = WfS[nt * 16 + col];
    v8f g;
#pragma unroll
    for (int i = 0; i < 8; ++i) g[i] = (a3m[nt][i] > 0.0f) ? wf : 0.0f;
    stC(SA, col, half, nt, g);
  }

  // ---------- g2 = (g3 @ W3^T) * (a2_g > 0); mask re-read from SG2 ----------
  layerT(SA, W3S, col, half, cg);
#pragma unroll
  for (int nt = 0; nt < 4; ++nt) {
    v8f m = ldC(SG2, col, half, nt);
#pragma unroll
    for (int i = 0; i < 8; ++i)
      cg[nt][i] = (m[i] > 0.0f) ? cg[nt][i] : 0.0f;
    stC(SG2, col, half, nt, cg[nt]);  // reads precede writes: in-order LDS
  }

  // ---------- g1 = (g2 @ W2^T) * (a1_g > 0); mask re-read from SG1 ----------
  layerT(SG2, W2S, col, half, cg);
#pragma unroll
  for (int nt = 0; nt < 4; ++nt) {
    v8f m = ldC(SG1, col, half, nt);
#pragma unroll
    for (int i = 0; i < 8; ++i)
      cg[nt][i] = (m[i] > 0.0f) ? cg[nt][i] : 0.0f;
    stC(SA, col, half, nt, cg[nt]);
  }

  // ---------- dy = g1 @ W1x^T  (B cols 0..3 = W1 rows 0..3, rest zero) ----------
  v8f dyc = {0, 0, 0, 0, 0, 0, 0, 0};
#pragma unroll
  for (int kc = 0; kc < 16; ++kc) {
    v2f a = ldA(SA, col, half, kc);
    const int k = kc * 4 + half * 2;
    v2f br = *(const v2f*)(W1S + (col & 3) * 64 + k);  // W1[j][k], contiguous
    v2f bt;
    bt[0] = (col < 4) ? br[0] : 0.0f;
    bt[1] = (col < 4) ? br[1] : 0.0f;
    dyc = wmma4(a, bt, dyc);
  }
  if (col < 4) {
#pragma unroll
    for (int r = 0; r < 8; ++r) SA[(half * 8 + r) * 4 + col] = dyc[r];
  }
  {
    v2f d;
    d[0] = SA[lane * 2];
    d[1] = SA[lane * 2 + 1];
    *(v2f*)(outDY + sbase * 4 + lane * 2) = d;  // coalesced b64 stores
  }

  // ---------- ddy == 0 exactly (piecewise-linear ReLU net) ----------
  {
    v4f z = {0, 0, 0, 0};
    float* p = outDDY + sbase * 16 + lane * 8;
    *(v4f*)(p) = z;
    *(v4f*)(p + 4) = z;
  }
}

extern "C" void kernel_launch(void* const* d_in, const int* in_sizes, int n_in,
                              void* d_out, int out_size, void* d_ws,
                              size_t ws_size, hipStream_t stream) {
  (void)n_in;
  (void)out_size;
  (void)d_ws;
  (void)ws_size;
  const float* x1 = (const float*)d_in[0];
  const float* x2 = (const float*)d_in[1];
  // d_in[2] (x3) is mathematically unused: Hessian of a ReLU MLP is 0.
  const float* x4 = (const float*)d_in[3];
  const float* W1 = (const float*)d_in[4];
  const float* b1 = (const float*)d_in[5];
  const float* W2 = (const float*)d_in[6];
  const float* b2 = (const float*)d_in[7];
  const float* W3 = (const float*)d_in[8];
  const float* b3 = (const float*)d_in[9];
  const float* Wf = (const float*)d_in[10];
  const float* bf = (const float*)d_in[11];

  const long B = in_sizes[0] / 4;  // x1 is [B,4]
  float* out = (float*)d_out;
  float* outY = out;            // [B]
  float* outDY = out + B;       // [B,4]
  float* outDDY = out + 5 * B;  // [B,4,4]

  const int samplesPerBlock = 16 * WAVES_PER_BLOCK;
  const int blocks = (int)((B + samplesPerBlock - 1) / samplesPerBlock);
  idnn_pinn_kernel<<<blocks, WAVES_PER_BLOCK * 32, 0, stream>>>(
      x1, x2, x4, W1, b1, W2, b2, W3, b3, Wf, bf, outY, outDY, outDDY);
}